// GraphEncoder_15650860826889
// MI455X (gfx1250) — compile-verified
//
#include <hip/hip_runtime.h>
#include <hip/hip_bf16.h>

typedef __attribute__((ext_vector_type(2))) float v2f;
typedef __attribute__((ext_vector_type(8))) float v8f;

#define NEG_SLOPE 0.2f
#define BN_EPS 1e-5f

// ---------------------------------------------------------------- utilities

__global__ void k_zero(float* __restrict__ p, int n) {
  int i = blockIdx.x * blockDim.x + threadIdx.x;
  if (i < n) p[i] = 0.f;
}

// monotonic float<->uint encoding so atomicMax(uint) == float max
__device__ __forceinline__ unsigned enc_f32(float f) {
  unsigned u = __float_as_uint(f);
  return (u & 0x80000000u) ? ~u : (u | 0x80000000u);
}
__device__ __forceinline__ float dec_f32(unsigned u) {
  return __uint_as_float((u & 0x80000000u) ? (u & 0x7fffffffu) : ~u);
}

// ------------------------------------------------- degree / self-loop attrs

__global__ void k_deg_loop(const float* __restrict__ ea, const int* __restrict__ dst,
                           float* __restrict__ cnt, float* __restrict__ lsum, int E) {
  int e = blockIdx.x * blockDim.x + threadIdx.x;
  if (e >= E) return;
  int dv = dst[e];
  atomicAdd(&cnt[dv], 1.f);
  const float* er = ea + (size_t)e * 16;
  float* lr = lsum + (size_t)dv * 16;
#pragma unroll
  for (int j = 0; j < 16; ++j) atomicAdd(&lr[j], er[j]);
}

__global__ void k_loop_fin(float* __restrict__ lattr, const float* __restrict__ cnt, int N16) {
  int i = blockIdx.x * blockDim.x + threadIdx.x;
  if (i >= N16) return;
  lattr[i] = lattr[i] / fmaxf(cnt[i >> 4], 1.f);
}

// ------------------------------------------------------------- WMMA GEMM
// C[M x 64] = A[M x 64] @ W[64 x 64], M % 16 == 0.
// One wave -> 16 rows x 64 cols; V_WMMA_F32_16X16X4_F32, K swept 4 at a time.
// A frag: lane l holds M=l&15, K-pair (l>>4)*2.  B frag mirrors (lane->N).
// C/D frag: VGPR r -> row r + 8*(lane>>4), col = (lane&15) + 16*ntile.
__global__ void k_gemm64_wmma(const float* __restrict__ A,
                              const float* __restrict__ W,
                              float* __restrict__ C, int M) {
  int wave = (int)(((unsigned)blockIdx.x * blockDim.x + threadIdx.x) >> 5);
  int lane = (int)(threadIdx.x & 31u);
  int row0 = wave * 16;
  if (row0 >= M) return;                 // wave-uniform: EXEC all-ones inside
  int ln = lane & 15;
  int kh = (lane >> 4) << 1;             // 0 or 2
  v8f acc0 = {}, acc1 = {}, acc2 = {}, acc3 = {};
  const float* Arow = A + (size_t)(row0 + ln) * 64;
#pragma unroll
  for (int k = 0; k < 64; k += 4) {
    v2f a;
    a.x = Arow[k + kh + 0];
    a.y = Arow[k + kh + 1];
    const float* W0 = W + (size_t)(k + kh) * 64 + ln;
    v2f b0, b1, b2, b3;
    b0.x = W0[0];  b0.y = W0[64];
    b1.x = W0[16]; b1.y = W0[80];
    b2.x = W0[32]; b2.y = W0[96];
    b3.x = W0[48]; b3.y = W0[112];
    acc0 = __builtin_amdgcn_wmma_f32_16x16x4_f32(false, a, false, b0, (short)0, acc0, false, false);
    acc1 = __builtin_amdgcn_wmma_f32_16x16x4_f32(false, a, false, b1, (short)0, acc1, false, false);
    acc2 = __builtin_amdgcn_wmma_f32_16x16x4_f32(false, a, false, b2, (short)0, acc2, false, false);
    acc3 = __builtin_amdgcn_wmma_f32_16x16x4_f32(false, a, false, b3, (short)0, acc3, false, false);
  }
  int rbase = row0 + (lane >> 4) * 8;
#pragma unroll
  for (int r = 0; r < 8; ++r) {
    float* Cr = C + (size_t)(rbase + r) * 64 + ln;
    Cr[0]  = acc0[r];
    Cr[16] = acc1[r];
    Cr[32] = acc2[r];
    Cr[48] = acc3[r];
  }
}

// ------------------------------------------------------------- edge kernels
// wave32 per edge; lane handles features (lane) and (lane+32).

__global__ void k_edge_logits(const float* __restrict__ xl, const float* __restrict__ xr,
                              const int* __restrict__ src, const int* __restrict__ dst,
                              const float* __restrict__ ea, const float* __restrict__ lattr,
                              const float* __restrict__ We, const float* __restrict__ att,
                              float* __restrict__ s, unsigned* __restrict__ smax,
                              int E, int EA) {
  int e = (int)(((unsigned)blockIdx.x * blockDim.x + threadIdx.x) >> 5);
  int lane = (int)(threadIdx.x & 31u);
  if (e >= EA) return;
  int sv, dv; const float* er;
  if (e < E) { sv = src[e]; dv = dst[e]; er = ea + (size_t)e * 16; }
  else       { sv = e - E;  dv = sv;     er = lattr + (size_t)sv * 16; }
  float a16[16];
#pragma unroll
  for (int k = 0; k < 16; ++k) a16[k] = er[k];   // broadcast loads, L0-friendly
  float accv = 0.f;
#pragma unroll
  for (int t = 0; t < 2; ++t) {
    int j = lane + t * 32;
    float ee = 0.f;
#pragma unroll
    for (int k = 0; k < 16; ++k) ee = fmaf(a16[k], We[k * 64 + j], ee);  // on-the-fly lin_edge
    float m = xl[(size_t)sv * 64 + j] + xr[(size_t)dv * 64 + j] + ee;
    float lr = m > 0.f ? m : NEG_SLOPE * m;
    accv = fmaf(lr, att[j], accv);
  }
  for (int off = 16; off > 0; off >>= 1) accv += __shfl_down(accv, off, 32);
  if (lane == 0) {
    s[e] = accv;
    atomicMax(&smax[dv], enc_f32(accv));
  }
}

__global__ void k_edge_expdenom(float* __restrict__ s, const unsigned* __restrict__ smax,
                                float* __restrict__ denom, const int* __restrict__ dst,
                                int E, int EA) {
  int e = blockIdx.x * blockDim.x + threadIdx.x;
  if (e >= EA) return;
  int dv = (e < E) ? dst[e] : (e - E);
  float ev = __expf(s[e] - dec_f32(smax[dv]));
  s[e] = ev;
  atomicAdd(&denom[dv], ev);
}

__global__ void k_edge_scatter(const float* __restrict__ s, const float* __restrict__ denom,
                               const float* __restrict__ xl,
                               const int* __restrict__ src, const int* __restrict__ dst,
                               float* __restrict__ acc, int E, int EA) {
  int e = (int)(((unsigned)blockIdx.x * blockDim.x + threadIdx.x) >> 5);
  int lane = (int)(threadIdx.x & 31u);
  if (e >= EA) return;
  int sv, dv;
  if (e < E) { sv = src[e]; dv = dst[e]; } else { sv = e - E; dv = sv; }
  float alpha = s[e] / denom[dv];
  const float* xls = xl + (size_t)sv * 64;
  float* ad = acc + (size_t)dv * 64;
  atomicAdd(&ad[lane],      alpha * xls[lane]);
  atomicAdd(&ad[lane + 32], alpha * xls[lane + 32]);
}

// ------------------------------------------------------------- node epilogue

__global__ void k_bias_act(float* __restrict__ h, const float* __restrict__ bias,
                           int n64, int do_elu) {
  int i = blockIdx.x * blockDim.x + threadIdx.x;
  if (i >= n64) return;
  float v = h[i] + bias[i & 63];
  if (do_elu) v = v > 0.f ? v : expm1f(v);
  h[i] = v;
}

__global__ void k_bn_stats(const float* __restrict__ h, float* __restrict__ sum,
                           float* __restrict__ sq, int N) {
  __shared__ float ls[4 * 64];
  __shared__ float lq[4 * 64];
  int j = threadIdx.x & 63;
  int g = threadIdx.x >> 6;
  float s = 0.f, q = 0.f;
  for (int i = blockIdx.x * 4 + g; i < N; i += gridDim.x * 4) {
    float v = h[(size_t)i * 64 + j];
    s += v; q += v * v;
  }
  ls[g * 64 + j] = s; lq[g * 64 + j] = q;
  __syncthreads();
  if (g == 0) {
    s = ls[j] + ls[64 + j] + ls[128 + j] + ls[192 + j];
    q = lq[j] + lq[64 + j] + lq[128 + j] + lq[192 + j];
    atomicAdd(&sum[j], s);
    atomicAdd(&sq[j], q);
  }
}

__global__ void k_bn_final(const float* __restrict__ sum, const float* __restrict__ sq,
                           const float* __restrict__ g, const float* __restrict__ b,
                           float* __restrict__ ab, int N) {
  int j = threadIdx.x;   // 64 threads
  float inv = 1.f / (float)N;
  float mu = sum[j] * inv;
  float var = sq[j] * inv - mu * mu;
  float a = g[j] * rsqrtf(var + BN_EPS);
  ab[j] = a;
  ab[64 + j] = b[j] - mu * a;
}

__global__ void k_bn_apply(float* __restrict__ h, const float* __restrict__ ab,
                           int n64, int post_elu) {
  int i = blockIdx.x * blockDim.x + threadIdx.x;
  if (i >= n64) return;
  int j = i & 63;
  float v = h[i] * ab[j] + ab[64 + j];
  if (post_elu) v = v > 0.f ? v : expm1f(v);
  h[i] = v;
}

// ---------------------------------------------------------------- launcher

static inline int cdiv(long long a, int b) { return (int)((a + b - 1) / b); }

extern "C" void kernel_launch(void* const* d_in, const int* in_sizes, int n_in,
                              void* d_out, int out_size, void* d_ws, size_t ws_size,
                              hipStream_t stream) {
  const float* x     = (const float*)d_in[0];
  const int*   ei    = (const int*)d_in[1];     // (2,E) int32 (JAX x64 off)
  const float* eattr = (const float*)d_in[2];
  const float* Wl    = (const float*)d_in[3];   // (3,64,64)
  const float* Wr    = (const float*)d_in[4];   // (3,64,64)
  const float* We    = (const float*)d_in[5];   // (3,16,64)
  const float* att   = (const float*)d_in[6];   // (3,64)
  const float* bias  = (const float*)d_in[7];   // (3,64)
  const float* gamma = (const float*)d_in[8];   // (2,64)
  const float* beta  = (const float*)d_in[9];   // (2,64)

  const int N  = in_sizes[0] / 64;              // 50000
  const int E  = in_sizes[1] / 2;               // 800000
  const int EA = E + N;
  const int* src = ei;
  const int* dst = ei + E;

  // -------- workspace carve (floats) --------
  float* w = (float*)d_ws;
  float* cnt   = w;  w += N;
  float* lattr = w;  w += (size_t)N * 16;       // contiguous with cnt (one zero pass)
  float* xl    = w;  w += (size_t)N * 64;
  float* xr    = w;  w += (size_t)N * 64;
  float* h1    = w;  w += (size_t)N * 64;
  float* h2    = w;  w += (size_t)N * 64;
  float* sbuf  = w;  w += EA;
  unsigned* smax = (unsigned*)w; w += N;
  float* denom = w;  w += N;
  float* bnsum = w;  w += 64;                   // bnsum/bnsq contiguous
  float* bnsq  = w;  w += 64;
  float* bnab  = w;  w += 128;

  const int n64 = N * 64;

  auto zero = [&](float* p, int n) {
    k_zero<<<cdiv(n, 256), 256, 0, stream>>>(p, n);
  };

  // -------- precompute: degree counts + self-loop edge attrs (shared by layers)
  zero(cnt, N + N * 16);                        // cnt + lattr in one pass
  k_deg_loop<<<cdiv(E, 256), 256, 0, stream>>>(eattr, dst, cnt, lattr, E);
  k_loop_fin<<<cdiv(N * 16, 256), 256, 0, stream>>>(lattr, cnt, N * 16);

  const int gemm_blocks = cdiv((long long)(N / 16), 8);     // 8 waves/block
  const int edge_wave_blocks = cdiv((long long)EA * 32, 256);

  // mode 0: +bias,ELU then BN0 ; mode 1: +bias then BN1,ELU ; mode 2: +bias only
  auto run_layer = [&](const float* hin, float* out, int l, int mode) {
    k_gemm64_wmma<<<gemm_blocks, 256, 0, stream>>>(hin, Wl + (size_t)l * 4096, xl, N);
    k_gemm64_wmma<<<gemm_blocks, 256, 0, stream>>>(hin, Wr + (size_t)l * 4096, xr, N);
    zero(out, n64);
    zero((float*)smax, N);
    zero(denom, N);
    k_edge_logits<<<edge_wave_blocks, 256, 0, stream>>>(
        xl, xr, src, dst, eattr, lattr, We + (size_t)l * 1024, att + l * 64,
        sbuf, smax, E, EA);
    k_edge_expdenom<<<cdiv(EA, 256), 256, 0, stream>>>(sbuf, smax, denom, dst, E, EA);
    k_edge_scatter<<<edge_wave_blocks, 256, 0, stream>>>(sbuf, denom, xl, src, dst, out, E, EA);
    if (mode == 0) {
      k_bias_act<<<cdiv(n64, 256), 256, 0, stream>>>(out, bias + l * 64, n64, 1);
      zero(bnsum, 128);
      k_bn_stats<<<128, 256, 0, stream>>>(out, bnsum, bnsq, N);
      k_bn_final<<<1, 64, 0, stream>>>(bnsum, bnsq, gamma, beta, bnab, N);
      k_bn_apply<<<cdiv(n64, 256), 256, 0, stream>>>(out, bnab, n64, 0);
    } else if (mode == 1) {
      k_bias_act<<<cdiv(n64, 256), 256, 0, stream>>>(out, bias + l * 64, n64, 0);
      zero(bnsum, 128);
      k_bn_stats<<<128, 256, 0, stream>>>(out, bnsum, bnsq, N);
      k_bn_final<<<1, 64, 0, stream>>>(bnsum, bnsq, gamma + 64, beta + 64, bnab, N);
      k_bn_apply<<<cdiv(n64, 256), 256, 0, stream>>>(out, bnab, n64, 1);
    } else {
      k_bias_act<<<cdiv(n64, 256), 256, 0, stream>>>(out, bias + l * 64, n64, 0);
    }
  };

  run_layer(x,  h1, 0, 0);
  run_layer(h1, h2, 1, 1);
  run_layer(h2, (float*)d_out, 2, 2);
}